// ReadoutPhase_773094113317
// MI455X (gfx1250) — compile-verified
//
#include <hip/hip_runtime.h>

// =====================================================================
// ReadoutPhase for MI455X (gfx1250, wave32)
//
// score = sigmoid(x @ W^T + b)           (GEMV, done via V_WMMA_F32_16X16X4_F32)
// out[:,0:128]  = segment_sum(score * x) (sorted batch -> register accumulators
// out[:,128:256]= segment_max(x)          + atomic flush at segment boundaries)
//
// Memory-bound: 512 MB of x @ 23.3 TB/s ~= 22 us floor. x is read once from
// HBM per tile; the second (dim-major) pass over the same 8 KB tile hits cache.
// =====================================================================

typedef float v2f __attribute__((ext_vector_type(2)));
typedef float v8f __attribute__((ext_vector_type(8)));

#define DIM 128
#define TILE 16            // nodes per WMMA tile (M dimension)
#define TILES_PER_WAVE 4   // 64 contiguous nodes per wave -> segment locality

__global__ void init_out_kernel(float* __restrict__ out, int out_size) {
  int idx = blockIdx.x * blockDim.x + threadIdx.x;
  if (idx < out_size) {
    // layout per graph: [0..127] = gated sum (identity 0), [128..255] = max (identity -inf)
    out[idx] = ((idx & 255) < DIM) ? 0.0f : -__builtin_inff();
  }
}

__device__ __forceinline__ void atomicMaxF(float* p, float v) {
  // lowers to global_atomic_max_num_f32 on gfx1250
  __hip_atomic_fetch_max(p, v, __ATOMIC_RELAXED, __HIP_MEMORY_SCOPE_AGENT);
}

__global__ void __launch_bounds__(256)
readout_kernel(const float* __restrict__ x,
               const long long* __restrict__ batch,
               const float* __restrict__ W,
               const float* __restrict__ b,
               float* __restrict__ out,
               int n_nodes) {
  const int lane = threadIdx.x & 31;
  const int wave = (blockIdx.x * blockDim.x + threadIdx.x) >> 5;
  const int half = lane >> 4;   // A-fragment K-pair select (K0/K1 vs K2/K3)
  const int rowm = lane & 15;   // A-fragment row M = lane % 16
  const float bias = b[0];

  const long long nodeBase = (long long)wave * (TILE * TILES_PER_WAVE);

  float4 accSum = make_float4(0.f, 0.f, 0.f, 0.f);
  float4 accMax = make_float4(-__builtin_inff(), -__builtin_inff(),
                              -__builtin_inff(), -__builtin_inff());
  int curSeg = -1;

  for (int it = 0; it < TILES_PER_WAVE; ++it) {
    const long long tb = nodeBase + (long long)it * TILE;
    if (tb >= n_nodes) break;                     // wave-uniform exit (EXEC stays full)

    // prefetch next tile (global_prefetch_b8)
    if (tb + TILE < n_nodes)
      __builtin_prefetch(x + (tb + TILE) * DIM + lane * 4, 0, 1);

    // ---- Phase 1: D[m,n] = dot(x[tb+m,:], W) for all n, via 32x WMMA f32 16x16x4.
    // A layout (ISA 7.12.2, 32-bit A 16x4): lanes 0-15 hold (K=4t, 4t+1),
    // lanes 16-31 hold (K=4t+2, 4t+3), row M = lane%16.
    // B = W broadcast into all 16 columns => every column of D equals the dot.
    v8f d = {0.f, 0.f, 0.f, 0.f, 0.f, 0.f, 0.f, 0.f};
    const float* xrow = x + (tb + rowm) * DIM + half * 2;
    const float* wp   = W + half * 2;
#pragma unroll
    for (int t = 0; t < DIM / 4; ++t) {
      float2 a2 = *(const float2*)(xrow + 4 * t);
      float2 w2 = *(const float2*)(wp   + 4 * t);
      v2f av; av[0] = a2.x; av[1] = a2.y;
      v2f bv; bv[0] = w2.x; bv[1] = w2.y;
      d = __builtin_amdgcn_wmma_f32_16x16x4_f32(false, av, false, bv,
                                                (short)0, d, false, false);
    }

    // ---- Phase 2: dim-major pass. Lane l owns dims [4l, 4l+4). Tile is cache-hot.
    const int mEnd = (n_nodes - tb >= TILE) ? TILE : (int)(n_nodes - tb);
    const float* xcol = x + tb * DIM + lane * 4;
#pragma unroll
    for (int m = 0; m < TILE; ++m) {
      if (m >= mEnd) break;
      const int seg = (int)batch[tb + m];        // sorted; scalar-uniform load
      if (seg != curSeg) {
        if (curSeg >= 0) {                       // flush previous segment
          float* os = out + (size_t)curSeg * (2 * DIM) + lane * 4;
          atomicAdd(os + 0, accSum.x);
          atomicAdd(os + 1, accSum.y);
          atomicAdd(os + 2, accSum.z);
          atomicAdd(os + 3, accSum.w);
          atomicMaxF(os + DIM + 0, accMax.x);
          atomicMaxF(os + DIM + 1, accMax.y);
          atomicMaxF(os + DIM + 2, accMax.z);
          atomicMaxF(os + DIM + 3, accMax.w);
        }
        curSeg = seg;
        accSum = make_float4(0.f, 0.f, 0.f, 0.f);
        accMax = make_float4(-__builtin_inff(), -__builtin_inff(),
                             -__builtin_inff(), -__builtin_inff());
      }
      // D element (m, n): VGPR m&7, lanes 0-15 for m<8 / lanes 16-31 for m>=8
      float dot = __shfl(d[m & 7], (m >> 3) << 4, 32);
      float sc  = 1.0f / (1.0f + __expf(-(dot + bias)));
      float4 xv = *(const float4*)(xcol + m * DIM);
      accSum.x += sc * xv.x;  accSum.y += sc * xv.y;
      accSum.z += sc * xv.z;  accSum.w += sc * xv.w;
      accMax.x = fmaxf(accMax.x, xv.x);  accMax.y = fmaxf(accMax.y, xv.y);
      accMax.z = fmaxf(accMax.z, xv.z);  accMax.w = fmaxf(accMax.w, xv.w);
    }
  }

  // final flush
  if (curSeg >= 0) {
    float* os = out + (size_t)curSeg * (2 * DIM) + lane * 4;
    atomicAdd(os + 0, accSum.x);
    atomicAdd(os + 1, accSum.y);
    atomicAdd(os + 2, accSum.z);
    atomicAdd(os + 3, accSum.w);
    atomicMaxF(os + DIM + 0, accMax.x);
    atomicMaxF(os + DIM + 1, accMax.y);
    atomicMaxF(os + DIM + 2, accMax.z);
    atomicMaxF(os + DIM + 3, accMax.w);
  }
}

extern "C" void kernel_launch(void* const* d_in, const int* in_sizes, int n_in,
                              void* d_out, int out_size, void* d_ws, size_t ws_size,
                              hipStream_t stream) {
  const float*     x     = (const float*)d_in[0];
  const long long* batch = (const long long*)d_in[1];
  const float*     W     = (const float*)d_in[2];
  const float*     b     = (const float*)d_in[3];
  float*           out   = (float*)d_out;

  const int n_nodes = in_sizes[0] / DIM;   // N = 1,000,000

  // Kernel 0: (re)initialize output identities every call (graph-replay safe).
  {
    int threads = 256;
    int blocks  = (out_size + threads - 1) / threads;
    init_out_kernel<<<blocks, threads, 0, stream>>>(out, out_size);
  }

  // Kernel 1: one wave per 64 contiguous nodes.
  {
    int waves   = (n_nodes + TILE * TILES_PER_WAVE - 1) / (TILE * TILES_PER_WAVE);
    int blocks  = (waves + 7) / 8;           // 8 waves (256 threads) per block
    readout_kernel<<<blocks, 256, 0, stream>>>(x, batch, W, b, out, n_nodes);
  }
}